// HyBRiDConstructor_51436528337257
// MI455X (gfx1250) — compile-verified
//
#include <hip/hip_runtime.h>
#include <hip/hip_bf16.h>

typedef __attribute__((ext_vector_type(2))) float v2f;
typedef __attribute__((ext_vector_type(8))) float v8f;

namespace {
constexpr int kG = 64;
constexpr int kN = 2048;
constexpr int kR = 16;
constexpr int kD = 256;
constexpr float kEps = 1e-6f;
// Pad LDS rows: 2048*4B stride is 0 mod 256B -> 16-way bank conflict on the
// A-fragment loads. 2052 floats => lane m lands on banks (4m..4m+1) mod 64.
constexpr int kLdsStride = kN + 4;  // floats
}

// ---------------------------------------------------------------------------
// Kernel 1: Gumbel-softmax masks. For 2 classes the whole
// softmax(log_softmax(l)+g)[...,1] collapses to
//   sigmoid((l1-l0) + log(-log u0) - log(-log u1)).
// One block per (g,r) row of N=2048. Also emits logits_out and batch_hyper.
// ---------------------------------------------------------------------------
__global__ __launch_bounds__(256) void hybrid_masks_kernel(
    const float* __restrict__ u,            // [G,R,N,2]
    const float* __restrict__ mask_logits,  // [R,N,2]
    float* __restrict__ masks_out,          // [G,R,N]
    float* __restrict__ logits_out,         // [G,R,N]
    int*   __restrict__ batch_hyper)        // [G*R]
{
    const int b   = blockIdx.x;          // g*R + r
    const int r   = b & (kR - 1);
    const int tid = threadIdx.x;

    const float2* __restrict__ urow  = (const float2*)(u + (size_t)b * kN * 2);
    const float2* __restrict__ mlrow = (const float2*)(mask_logits + (size_t)r * kN * 2);
    float* __restrict__ mrow = masks_out  + (size_t)b * kN;
    float* __restrict__ lrow = logits_out + (size_t)b * kN;

    #pragma unroll
    for (int n = tid; n < kN; n += 256) {
        const float2 uv = urow[n];
        const float2 ml = mlrow[n];
        const float l0 = -__logf(uv.x);          // -log(u0) > 0
        const float l1 = -__logf(uv.y);          // -log(u1) > 0
        const float z  = (ml.y - ml.x) + __logf(l0) - __logf(l1);
        const float m  = 1.0f / (1.0f + __expf(-z));
        mrow[n] = m;
        lrow[n] = ml.y;
    }
    if (tid == 0) batch_hyper[b] = b >> 4;       // = g
}

// ---------------------------------------------------------------------------
// Kernel 2: per-graph [16 x 2048] @ [2048 x 256] via V_WMMA_F32_16X16X4_F32.
// Grid (G, 4): block owns 64 output columns; 4 waves, one 16-col tile each.
// A (masks) staged in bank-conflict-free LDS; B (x) streamed from HBM once.
// Denominators recomputed from the LDS copy (no extra global traffic).
// ---------------------------------------------------------------------------
__global__ __launch_bounds__(128) void hybrid_gemm_kernel(
    const float* __restrict__ x,       // [G*N, D]
    const float* __restrict__ masks,   // [G,R,N] (written by kernel 1)
    float* __restrict__ h_out)         // [G*R, D]
{
    extern __shared__ float smem[];
    float* amat = smem;                      // [kR][kLdsStride]
    float* dsh  = smem + kR * kLdsStride;    // [kR] denominators

    const int g    = blockIdx.x;
    const int cb   = blockIdx.y;             // 64-column chunk 0..3
    const int tid  = threadIdx.x;
    const int wave = tid >> 5;               // 0..3
    const int lane = tid & 31;

    if (tid < kR) dsh[tid] = kEps;

    // Stage masks[g] (128 KB) into LDS with padded stride.
    const float4* __restrict__ msrc = (const float4*)(masks + (size_t)g * kR * kN);
    for (int i = tid; i < kR * (kN / 4); i += 128) {
        const int row = i >> 9;              // / 512 float4s per row
        const int c4  = i & 511;
        const float4 v = msrc[i];
        float* dst = &amat[row * kLdsStride + c4 * 4];
        dst[0] = v.x; dst[1] = v.y; dst[2] = v.z; dst[3] = v.w;
    }
    __syncthreads();

    // Row sums of masks: 8 threads per row, 256 elements each, LDS atomics.
    {
        const int row = tid >> 3;
        const int seg = tid & 7;
        const float* p = &amat[row * kLdsStride + seg * (kN / 8)];
        float s = 0.f;
        #pragma unroll 8
        for (int i = 0; i < kN / 8; ++i) s += p[i];
        atomicAdd(&dsh[row], s);
    }
    __syncthreads();

    // WMMA fragment addressing (ISA 7.12.2, f32 16x16x4):
    //  A: lane m (=lane%16) holds row m; K = {0,1} low half, {2,3} high half.
    //  B: VGPR0 = rows k0 / k0+2 striped over lanes (N = lane%16), VGPR1 = +1.
    const int colbase = cb * 64 + wave * 16;
    const int nlo  = lane & 15;
    const int koff = (lane >> 4) << 1;       // 0 or 2
    const float* __restrict__ xg   = x + (size_t)g * kN * kD + colbase + nlo;
    const float* __restrict__ arow = &amat[nlo * kLdsStride + koff];

    v8f c = {0.f, 0.f, 0.f, 0.f, 0.f, 0.f, 0.f, 0.f};
    for (int kk = 0; kk < kN; kk += 64) {
        // keep the x stream ahead in L2 (global_prefetch_b8); wrap keeps it in-bounds
        __builtin_prefetch(xg + (size_t)((kk + 128) & (kN - 1)) * kD, 0, 1);
        #pragma unroll
        for (int k0 = kk; k0 < kk + 64; k0 += 4) {
            const v2f a = *(const v2f*)(arow + k0);
            v2f bt;
            bt.x = xg[(size_t)(k0 + koff) * kD];
            bt.y = xg[(size_t)(k0 + koff + 1) * kD];
            c = __builtin_amdgcn_wmma_f32_16x16x4_f32(
                    false, a, false, bt, (short)0, c, false, false);
        }
    }

    // D layout: VGPR j -> row j (lanes 0-15) or j+8 (lanes 16-31), col = lane%16.
    const int rbase = (lane >> 4) * 8;
    float* __restrict__ hp = h_out + (size_t)g * kR * kD + colbase + nlo;
    #pragma unroll
    for (int j = 0; j < 8; ++j) {
        const int rr = rbase + j;
        hp[(size_t)rr * kD] = c[j] / dsh[rr];
    }
}

// ---------------------------------------------------------------------------
extern "C" void kernel_launch(void* const* d_in, const int* in_sizes, int n_in,
                              void* d_out, int out_size, void* d_ws, size_t ws_size,
                              hipStream_t stream) {
    const float* x  = (const float*)d_in[0];  // [G*N, D]
    const float* ml = (const float*)d_in[1];  // [R, N, 2]
    // d_in[2] = batch ids: unused (sorted layout is implicit in the indexing)
    const float* u  = (const float*)d_in[3];  // [G, R, N, 2]

    float* out = (float*)d_out;
    float* h_out      = out;                               // [G*R, D]
    float* masks_out  = out + (size_t)kG * kR * kD;        // [G, R, N]
    float* logits_out = masks_out + (size_t)kG * kR * kN;  // [G, R, N]
    int*   batch_hyp  = (int*)(logits_out + (size_t)kG * kR * kN);  // [G*R]

    hybrid_masks_kernel<<<kG * kR, 256, 0, stream>>>(u, ml, masks_out,
                                                     logits_out, batch_hyp);

    dim3 grid(kG, kD / 64);
    const size_t shmem = (size_t)(kR * kLdsStride + kR) * sizeof(float);
    hybrid_gemm_kernel<<<grid, 128, shmem, stream>>>(x, masks_out, h_out);
}